// LPR_68058051772597
// MI455X (gfx1250) — compile-verified
//
#include <hip/hip_runtime.h>
#include <math.h>

#define KNB 16
#define BLK 256
#define MAXSEG 16
#define ROWW (KNB + 1)   // LDS row stride in ints: 17 (odd -> conflict-free on 64 banks)

// ---------------------------------------------------------------------------
// Kernel 0: init workspace (segment maxima = 0, cumsum of lengths)
// ---------------------------------------------------------------------------
__global__ __launch_bounds__(64) void lpr_init(unsigned* __restrict__ segmax,
                                               int* __restrict__ cums,
                                               const int* __restrict__ lengths,
                                               int b) {
  int t = threadIdx.x;
  if (t < MAXSEG) segmax[t] = 0u;
  if (t == 0) {
    int acc = 0;
    for (int s = 0; s < b; ++s) { acc += lengths[s]; cums[s] = acc; }
  }
}

// ---------------------------------------------------------------------------
// Kernel 1: per-segment max of squared global distance (float bits as uint,
// valid ordering since values are >= 0). LDS pre-reduction, then one global
// atomic per (block, segment). Deterministic: max is order-independent.
// ---------------------------------------------------------------------------
__global__ __launch_bounds__(BLK) void lpr_segmax(const float* __restrict__ xyz,
                                                  const int* __restrict__ cums,
                                                  unsigned* __restrict__ segmax,
                                                  int n, int b) {
  __shared__ unsigned smax[MAXSEG];
  int t = threadIdx.x;
  if (t < MAXSEG) smax[t] = 0u;
  __syncthreads();

  int i = blockIdx.x * BLK + t;
  if (i < n) {
    float x = xyz[3 * (size_t)i + 0];
    float y = xyz[3 * (size_t)i + 1];
    float z = xyz[3 * (size_t)i + 2];
    float gd2 = x * x + y * y + z * z;
    int seg = 0;
    for (int s = 0; s < b; ++s) seg += (i >= cums[s]) ? 1 : 0;
    if (seg >= b) seg = b - 1;
    atomicMax(&smax[seg], __float_as_uint(gd2));
  }
  __syncthreads();
  if (t < b) {
    unsigned v = smax[t];
    if (v) atomicMax(&segmax[t], v);
  }
}

// ---------------------------------------------------------------------------
// Kernel 2: main per-point kernel.
//   - async-stage this thread's 16 neighbor indices into LDS (CDNA5
//     global_load_async_to_lds_b32, tracked by ASYNCcnt), overlapped with the
//     center-point load and segment/global-volume resolution
//   - gather neighbors (L2-resident), centroid -> direction angles ONCE/point
//   - per-neighbor: angles, distance, exp; write 9-ch rows + geo + ratio
// ---------------------------------------------------------------------------
__global__ __launch_bounds__(BLK) void lpr_main(const float* __restrict__ xyz,
                                                const int* __restrict__ nidx,
                                                const int* __restrict__ cums,
                                                const unsigned* __restrict__ segmax,
                                                float* __restrict__ out_rep,
                                                float* __restrict__ out_geo,
                                                float* __restrict__ out_ratio,
                                                int n, int b) {
  __shared__ int s_idx[BLK * ROWW];
  const int t = threadIdx.x;
  const int i = blockIdx.x * BLK + t;
  const bool valid = (i < n);

  // LDS byte offset of this thread's row (flat shared pointer low 32 bits ==
  // allocation-relative LDS offset; HW adds LDS_BASE).
  unsigned lds_row = (unsigned)(size_t)(&s_idx[t * ROWW]);

  if (valid) {
    unsigned goff = (unsigned)i * (KNB * 4u);  // byte offset into neigh_idx
#pragma unroll
    for (int k = 0; k < KNB; ++k) {
      // GVS form: loads from the 64b SGPR base in saddr plus the 32b lane offset in vaddr to the LDS address in vdst
      asm volatile("global_load_async_to_lds_b32 %0, %1, %2"
                   :: "v"(lds_row + 4u * (unsigned)k),
                      "v"(goff + 4u * (unsigned)k),
                      "s"(nidx)
                   : "memory");
    }
  }

  // ---- independent work overlapped with the async copy ----
  float cx = 0.f, cy = 0.f, cz = 0.f, gvol = 1.f;
  if (valid) {
    cx = xyz[3 * (size_t)i + 0];
    cy = xyz[3 * (size_t)i + 1];
    cz = xyz[3 * (size_t)i + 2];
    int seg = 0;
    for (int s = 0; s < b; ++s) seg += (i >= cums[s]) ? 1 : 0;
    if (seg >= b) seg = b - 1;
    gvol = sqrtf(__uint_as_float(segmax[seg]));  // sqrt(max d^2) == max sqrt(d^2)
  }

  // Wait for this wave's async LDS writes (own-row consumption: no barrier).
  asm volatile("s_wait_asynccnt 0x0" ::: "memory");

  if (!valid) return;

  float nbx[KNB], nby[KNB], nbz[KNB];
  float sx = 0.f, sy = 0.f, sz = 0.f;
#pragma unroll
  for (int k = 0; k < KNB; ++k) {
    int j = s_idx[t * ROWW + k];
    float vx = 0.f, vy = 0.f, vz = 0.f;
    if ((unsigned)j < (unsigned)n) {           // pad index n -> zero row
      const float* p = xyz + 3 * (size_t)j;
      vx = p[0]; vy = p[1]; vz = p[2];
    }
    nbx[k] = vx; nby[k] = vy; nbz[k] = vz;
    sx += vx; sy += vy; sz += vz;
  }

  const float rk = 1.0f / (float)KNB;
  float dx = cx - sx * rk, dy = cy - sy * rk, dz = cz - sz * rk;
  float da = atan2f(dy, dx);                              // once per point
  float db = atan2f(dz, sqrtf(dx * dx + dy * dy));        // once per point

  float* rep = out_rep + (size_t)i * (KNB * 9);
  float* geo = out_geo + (size_t)i * KNB;
  float maxdis = 0.f;
#pragma unroll
  for (int k = 0; k < KNB; ++k) {
    float rx = cx - nbx[k], ry = cy - nby[k], rz = cz - nbz[k];
    float xy2 = rx * rx + ry * ry;
    float dis = sqrtf(xy2 + rz * rz);
    float ra  = atan2f(ry, rx) - da;
    float rb  = atan2f(rz, sqrtf(xy2)) - db;
    maxdis = fmaxf(maxdis, dis);
    float* o = rep + k * 9;
    o[0] = ra;  o[1] = rb;  o[2] = dis;
    o[3] = cx;  o[4] = cy;  o[5] = cz;
    o[6] = nbx[k]; o[7] = nby[k]; o[8] = nbz[k];
    geo[k] = expf(-dis);
  }
  out_ratio[i] = (maxdis * maxdis * maxdis) / gvol;
}

// ---------------------------------------------------------------------------
extern "C" void kernel_launch(void* const* d_in, const int* in_sizes, int n_in,
                              void* d_out, int out_size, void* d_ws, size_t ws_size,
                              hipStream_t stream) {
  const float* xyz     = (const float*)d_in[0];
  const int*   nidx    = (const int*)d_in[1];
  const int*   lengths = (const int*)d_in[2];
  int n = in_sizes[0] / 3;
  int b = in_sizes[2];
  if (b > MAXSEG) b = MAXSEG;

  unsigned* segmax = (unsigned*)d_ws;
  int*      cums   = (int*)d_ws + MAXSEG;

  float* out_rep   = (float*)d_out;
  float* out_geo   = out_rep + (size_t)n * KNB * 9;
  float* out_ratio = out_geo + (size_t)n * KNB;

  int blocks = (n + BLK - 1) / BLK;
  lpr_init  <<<1, 64, 0, stream>>>(segmax, cums, lengths, b);
  lpr_segmax<<<blocks, BLK, 0, stream>>>(xyz, cums, segmax, n, b);
  lpr_main  <<<blocks, BLK, 0, stream>>>(xyz, nidx, cums, segmax,
                                         out_rep, out_geo, out_ratio, n, b);
}